// BEV_KNN_83940840833744
// MI455X (gfx1250) — compile-verified
//
#include <hip/hip_runtime.h>

// KNN post-processing (RangeNet++-style) for MI455X / gfx1250.
// Latency/VALU-bound random-gather kernel. Branchless: 26 neighborhood
// gathers issue as one clause (max MLP); in-bounds predicates carried as
// -1/0 integer masks used with bitwise AND. Top-4 (center is free) found by a
// single-scan branchless sorted-insertion network (no poison passes).
// Volume (64 MB) is L2-resident; streams use non-temporal hints.

constexpr int   kKNN     = 5;
constexpr int   kD       = 32;
constexpr int   kH       = 128;
constexpr int   kW       = 2048;
constexpr int   kClasses = 21;            // NCLASSES; invalid bucket == kClasses
constexpr float kCutoff  = 1.0f;
constexpr int   kSlice   = kH * kW;       // 262144 elements per z-slice

__device__ __forceinline__ float mask_f32(float v, int m) {
  return __int_as_float(__float_as_int(v) & m);   // m is -1 (keep) or 0 (zero)
}

__global__ __launch_bounds__(256) void bev_knn_kernel(
    const float* __restrict__ proj_range,
    const float* __restrict__ unproj_range,
    const int*   __restrict__ proj_argmax,
    const int*   __restrict__ px,
    const int*   __restrict__ py,
    const int*   __restrict__ pz,
    int*         __restrict__ out,
    int n)
{
  const int p = blockIdx.x * 256 + threadIdx.x;
  if (p >= n) return;

  // Streaming inputs: non-temporal so they don't evict the L2-resident volume.
  const float ur = __builtin_nontemporal_load(unproj_range + p);
  const int   x  = __builtin_nontemporal_load(px + p);
  const int   y  = __builtin_nontemporal_load(py + p);
  const int   z  = __builtin_nontemporal_load(pz + p);

  const int cbase = (z * kH + y) * kW + x;        // center flat index (< 2^23)

  // 2+2+2 shared bounds predicates (kept as lane masks, combined on SALU).
  const bool zlo = z >= 1, zhi = z < kD - 1;
  const bool ylo = y >= 1, yhi = y < kH - 1;
  const bool xlo = x >= 1, xhi = x < kW - 1;

  // ---- phase 1: per-neighbor -1/0 mask + safe flat index (OOB -> 0) ----
  int okm[27];
  int sidx[27];
#pragma unroll
  for (int j = 0; j < 27; ++j) {
    if (j == 13) continue;                         // center: no load needed
    const int dz = j / 9, dy = (j / 3) % 3, dx = j % 3;          // 0..2
    bool ok = true;
    if (dz == 0) ok = ok && zlo; else if (dz == 2) ok = ok && zhi;
    if (dy == 0) ok = ok && ylo; else if (dy == 2) ok = ok && yhi;
    if (dx == 0) ok = ok && xlo; else if (dx == 2) ok = ok && xhi;
    const int off = (dz - 1) * kSlice + (dy - 1) * kW + (dx - 1);
    okm[j]  = ok ? -1 : 0;                         // one cndmask
    sidx[j] = (cbase + off) & okm[j];              // and -> always-valid index
  }

  // ---- phase 2: all 26 gathers back-to-back (single clause, max MLP) ----
  float rv[27];
#pragma unroll
  for (int j = 0; j < 27; ++j) {
    if (j == 13) continue;
    rv[j] = proj_range[sidx[j]];
  }
  // Center class: always in-bounds, always selected (dist == 0); issue early
  // so its latency overlaps the selection ALU work.
  const int ccls = proj_argmax[cbase];

  // ---- phase 3+4 fused: distance, then single-scan sorted-insertion top-4.
  // Center (dist 0, index 13) is always in the smallest-5 set; taken free.
  // Sorted invariant (b0,j0) <= ... <= (b3,j3) by (dist, idx). Strict '<'
  // compares + ascending-index scan reproduce lax.top_k tie ordering exactly.
  // Compare bits are monotone (c0<=c1<=c2<=c3), so the shift network is exact.
  float b0 = __builtin_inff(), b1 = __builtin_inff();
  float b2 = __builtin_inff(), b3 = __builtin_inff();
  int   j0 = 0, j1 = 0, j2 = 0, j3 = 0;
#pragma unroll
  for (int j = 0; j < 27; ++j) {
    if (j == 13) continue;
    const float r = mask_f32(rv[j], okm[j]);       // bitwise zero-fill (pad=0)
    const float ad = fabsf(r - ur);
    const float e = (r < 0.0f) ? __builtin_inff() : ad;   // r<0 -> inf

    const bool c0 = e < b0;
    const bool c1 = e < b1;
    const bool c2 = e < b2;
    const bool c3 = e < b3;

    const float t1 = c0 ? b0 : e;                  // shifted-down values
    const float t2 = c1 ? b1 : e;
    const float t3 = c2 ? b2 : e;
    const int   u1 = c0 ? j0 : j;                  // shifted-down indices
    const int   u2 = c1 ? j1 : j;
    const int   u3 = c2 ? j2 : j;

    b3 = c3 ? t3 : b3;   j3 = c3 ? u3 : j3;
    b2 = c2 ? t2 : b2;   j2 = c2 ? u2 : j2;
    b1 = c1 ? t1 : b1;   j1 = c1 ? u1 : j1;
    b0 = c0 ? e  : b0;   j0 = c0 ? j  : j0;
  }

  const int   sel[kKNN]   = { 13,   j0, j1, j2, j3 };
  const float sdist[kKNN] = { 0.0f, b0, b1, b2, b3 };

  // ---- phase 5: class gathers for the 4 non-center picks (clustered) ----
  int caddr[kKNN];
  int cmask[kKNN];
#pragma unroll
  for (int k = 1; k < kKNN; ++k) {
    const int bj = sel[k];
    const int zz = z + (bj / 9) - 1;
    const int yy = y + ((bj / 3) % 3) - 1;
    const int xx = x + (bj % 3) - 1;
    const bool ok = ((unsigned)zz < (unsigned)kD) &
                    ((unsigned)yy < (unsigned)kH) &
                    ((unsigned)xx < (unsigned)kW);
    cmask[k] = ok ? -1 : 0;
    caddr[k] = ((zz * kH + yy) * kW + xx) & cmask[k];
  }
  int craw[kKNN];
#pragma unroll
  for (int k = 1; k < kKNN; ++k) craw[k] = proj_argmax[caddr[k]];

  int cls[kKNN];
  cls[0] = ccls;                                   // center: never cut off
#pragma unroll
  for (int k = 1; k < kKNN; ++k) {
    const int c = craw[k] & cmask[k];              // OOB -> padding class 0
    cls[k] = (sdist[k] > kCutoff) ? kClasses : c;  // cutoff -> invalid bucket
  }

  // ---- phase 6: plurality vote over classes 1..kClasses-1.
  // argmax-first semantics: ties -> smallest class; all-zero -> class 1.
  int bestC = 1, bestCnt = 0;
#pragma unroll
  for (int i = 0; i < kKNN; ++i) {
    const int c = cls[i];
    int cnt = 0;
#pragma unroll
    for (int jj = 0; jj < kKNN; ++jj) cnt += (cls[jj] == c) ? 1 : 0;
    const bool valid = (c >= 1) & (c <= kClasses - 1);
    const bool take  = valid & ((cnt > bestCnt) | ((cnt == bestCnt) & (c < bestC)));
    bestCnt = take ? cnt : bestCnt;
    bestC   = take ? c   : bestC;
  }

  __builtin_nontemporal_store(bestC, out + p);
}

extern "C" void kernel_launch(void* const* d_in, const int* in_sizes, int n_in,
                              void* d_out, int out_size, void* d_ws, size_t ws_size,
                              hipStream_t stream) {
  const float* proj_range   = (const float*)d_in[0];
  const float* unproj_range = (const float*)d_in[1];
  const int*   proj_argmax  = (const int*)d_in[2];
  const int*   px           = (const int*)d_in[3];
  const int*   py           = (const int*)d_in[4];
  const int*   pz           = (const int*)d_in[5];
  int* out = (int*)d_out;

  const int n     = in_sizes[1];          // P = number of points
  const int block = 256;                  // 8 wave32 waves per block
  const int grid  = (n + block - 1) / block;
  bev_knn_kernel<<<grid, block, 0, stream>>>(proj_range, unproj_range,
                                             proj_argmax, px, py, pz, out, n);
}